// EfficientByteSHIFT_7945689497963
// MI455X (gfx1250) — compile-verified
//
#include <hip/hip_runtime.h>

// EfficientByteSHIFT for MI455X (gfx1250, wave32).
// Layout: [B,S,128] f32. One wave per token; lane l owns channels 4l..4l+3
// via one b128 load/store (fully coalesced 512B per token per direction).
// Memory-bound: 256 MiB total @ 23.3 TB/s ~= 11.5 us floor.

typedef float v4f __attribute__((ext_vector_type(4)));

__global__ __launch_bounds__(256) void EfficientByteSHIFT_gfx1250_kernel(
    const float* __restrict__ x, float* __restrict__ out, int nTok) {
  const int lane   = threadIdx.x & 31;
  const int wave   = (blockIdx.x * blockDim.x + threadIdx.x) >> 5;
  const int nWaves = (gridDim.x * blockDim.x) >> 5;

  for (int t = wave; t < nTok; t += nWaves) {
    const size_t base = (size_t)t * 128 + 4 * lane;
    // Streaming (non-temporal) 128-bit load of this lane's 4 channels.
    v4f f = __builtin_nontemporal_load((const v4f*)(x + base));

    // gfx1250 prefetch path (global_prefetch_b8) for the next token this
    // wave will process — one grid sweep ahead.
    const int tn = t + nWaves;
    if (tn < nTok)
      __builtin_prefetch(x + (size_t)tn * 128 + 4 * lane, 0, 0);

    // --- per-lane argmax over this lane's quad (prefer lowest index) ---
    float v = f[0]; int idx = 0;
    if (f[1] > v) { v = f[1]; idx = 1; }
    if (f[2] > v) { v = f[2]; idx = 2; }
    if (f[3] > v) { v = f[3]; idx = 3; }
    idx += (lane & 3) * 4;          // bank-local index 0..15 in each quad group

    // --- argmax reduction across each 4-lane group (max val, min idx tie) ---
    #pragma unroll
    for (int m = 1; m <= 2; m <<= 1) {
      const float vo = __shfl_xor(v,  m, 32);
      const int   io = __shfl_xor(idx, m, 32);
      if (vo > v || (vo == v && io < idx)) { v = vo; idx = io; }
    }
    // Now lanes 4-7 hold argmax(ALU_LO), 8-11 argmax(ALU_HI), 12-15 argmax(shift).

    // --- broadcast decoded fields to the whole wave ---
    const float mark_f = __shfl(f[0], 0, 32);   // ch 0
    const float shl_f  = __shfl(f[1], 0, 32);   // ch 1
    const float shr_f  = __shfl(f[2], 0, 32);   // ch 2
    const int val_lo   = __shfl(idx, 4, 32);    // argmax ch 16..31
    const int val_hi   = __shfl(idx, 8, 32);    // argmax ch 32..47
    int shift          = __shfl(idx, 12, 32);   // argmax ch 48..63
    shift = (shift > 31) ? 31 : shift;          // mirrors reference clamp

    const bool is_shl = shl_f > 0.5f;
    const bool is_shr = shr_f > 0.5f;
    const bool active = (mark_f >= 0.5f) && (is_shl || is_shr);

    const unsigned value   = (unsigned)val_lo | ((unsigned)val_hi << 4); // 0..255
    const unsigned shl_res = (value << shift) & 255u;
    const unsigned shr_res = value >> shift;
    const unsigned result  = is_shl ? shl_res : shr_res;

    const int ch_lo = 64 + (int)(result & 15u);   // 64..79
    const int ch_hi = 80 + (int)(result >> 4);    // 80..95 (never equals ch_lo)
    const float add = active ? 2.0f : 0.0f;

    const int c0 = 4 * lane;
    v4f g;
    g[0] = f[0] + ((c0 + 0 == ch_lo) ? add : 0.0f) + ((c0 + 0 == ch_hi) ? add : 0.0f);
    g[1] = f[1] + ((c0 + 1 == ch_lo) ? add : 0.0f) + ((c0 + 1 == ch_hi) ? add : 0.0f);
    g[2] = f[2] + ((c0 + 2 == ch_lo) ? add : 0.0f) + ((c0 + 2 == ch_hi) ? add : 0.0f);
    g[3] = f[3] + ((c0 + 3 == ch_lo) ? add : 0.0f) + ((c0 + 3 == ch_hi) ? add : 0.0f);

    // Streaming 128-bit store (output is written once, never re-read).
    __builtin_nontemporal_store(g, (v4f*)(out + base));
  }
}

extern "C" void kernel_launch(void* const* d_in, const int* in_sizes, int n_in,
                              void* d_out, int out_size, void* d_ws, size_t ws_size,
                              hipStream_t stream) {
  const float* x = (const float*)d_in[0];
  float* out     = (float*)d_out;
  const int nTok = in_sizes[0] / 128;   // B*S tokens, 128 channels each

  // 8 tokens per wave -> prefetch distance of one grid sweep; 256 threads =
  // 8 wave32s per workgroup (fills a WGP's 4 SIMD32s twice over).
  const int threads    = 256;
  const int totalWaves = (nTok + 7) / 8;
  int blocks = (totalWaves * 32 + threads - 1) / threads;
  if (blocks < 1) blocks = 1;

  EfficientByteSHIFT_gfx1250_kernel<<<blocks, threads, 0, stream>>>(x, out, nTok);
}